// GroupQueryAttention_13288628814302
// MI455X (gfx1250) — compile-verified
//
#include <hip/hip_runtime.h>

// ---------------------------------------------------------------------------
// GQA attention block for MI455X (gfx1250), fp32 via V_WMMA_F32_16X16X4_F32.
// ---------------------------------------------------------------------------

typedef __attribute__((ext_vector_type(2))) float v2f;
typedef __attribute__((ext_vector_type(8))) float v8f;

#define T_SEQ   4096
#define DIM_    1024
#define HEADS_  16
#define KVH_    4
#define HD_     64
#define KVDIM_  256
#define SCALE_  0.125f   // 64^-0.5

__device__ __forceinline__ v8f wmma_f32(v2f a, v2f b, v8f c) {
  // D = A(16x4) * B(4x16) + C(16x16), all fp32, wave32.
  return __builtin_amdgcn_wmma_f32_16x16x4_f32(
      /*neg_a=*/false, a, /*neg_b=*/false, b,
      /*c_mod=*/(short)0, c, /*reuse_a=*/false, /*reuse_b=*/false);
}

__device__ __forceinline__ v8f zero8() {
  v8f z = {0.f, 0.f, 0.f, 0.f, 0.f, 0.f, 0.f, 0.f};
  return z;
}

// ---------------------------------------------------------------------------
// GEMM: D = X[M,K] * W[N,K]^T + bias[N]
// MODE 0: D is [M,N] row-major.
// MODE 1: D is head-split [N/64, T_SEQ, 64]  (i.e. [Hk, T, hd] for K/V cache).
// Block: 256 threads (8 waves), macro tile 128(M) x 64(N), K-block 32.
// ---------------------------------------------------------------------------
template <int MODE>
__global__ __launch_bounds__(256) void gemm_kernel(
    const float* __restrict__ X, const float* __restrict__ W,
    const float* __restrict__ bias, float* __restrict__ D,
    int M, int N, int K) {
  __shared__ float As[128][36];  // stride 36: 16B-aligned fills, conflict-free frags
  __shared__ float Bs[64][36];

  const int tid   = threadIdx.x;
  const int lane  = tid & 31;
  const int wv    = tid >> 5;
  const int l15   = lane & 15;
  const int hoff  = (lane >> 4) << 1;  // K-offset within fragment: 0 or 2
  const int half8 = (lane >> 4) << 3;  // row offset in C layout: 0 or 8

  const int rowBase = blockIdx.x * 128;
  const int colBase = blockIdx.y * 64;
  const int mW = (wv >> 1) * 32;  // wave tile inside macro tile
  const int nW = (wv & 1) * 32;

  v8f c00 = zero8(), c01 = zero8(), c10 = zero8(), c11 = zero8();

  const int ar = tid >> 1, ac = (tid & 1) << 4;  // 2 thr/row, 16 floats each
  const int br = tid >> 2, bc = (tid & 3) << 3;  // 4 thr/row, 8 floats each

  for (int kb = 0; kb < K; kb += 32) {
    __syncthreads();
    const float* xp = X + (size_t)(rowBase + ar) * K + kb + ac;
#pragma unroll
    for (int j = 0; j < 4; ++j)
      *(float4*)&As[ar][ac + j * 4] = *(const float4*)(xp + j * 4);
    const float* wp = W + (size_t)(colBase + br) * K + kb + bc;
#pragma unroll
    for (int j = 0; j < 2; ++j)
      *(float4*)&Bs[br][bc + j * 4] = *(const float4*)(wp + j * 4);
    __syncthreads();

#pragma unroll
    for (int kd = 0; kd < 32; kd += 4) {
      v2f a0 = *(const v2f*)&As[mW + l15][kd + hoff];
      v2f a1 = *(const v2f*)&As[mW + 16 + l15][kd + hoff];
      v2f b0 = *(const v2f*)&Bs[nW + l15][kd + hoff];
      v2f b1 = *(const v2f*)&Bs[nW + 16 + l15][kd + hoff];
      c00 = wmma_f32(a0, b0, c00);
      c01 = wmma_f32(a0, b1, c01);
      c10 = wmma_f32(a1, b0, c10);
      c11 = wmma_f32(a1, b1, c11);
    }
  }

  const float bias0 = bias[colBase + nW + l15];
  const float bias1 = bias[colBase + nW + 16 + l15];
  const int col0 = colBase + nW + l15;
  const int col1 = colBase + nW + 16 + l15;
#pragma unroll
  for (int r = 0; r < 8; ++r) {
    const int row0 = rowBase + mW + half8 + r;       // c00 / c01 rows
    const int row1 = rowBase + mW + 16 + half8 + r;  // c10 / c11 rows
    float e00 = c00[r] + bias0, e01 = c01[r] + bias1;
    float e10 = c10[r] + bias0, e11 = c11[r] + bias1;
    if (MODE == 0) {
      D[(size_t)row0 * N + col0] = e00;
      D[(size_t)row0 * N + col1] = e01;
      D[(size_t)row1 * N + col0] = e10;
      D[(size_t)row1 * N + col1] = e11;
    } else {
      D[((size_t)(col0 >> 6) * T_SEQ + row0) * 64 + (col0 & 63)] = e00;
      D[((size_t)(col1 >> 6) * T_SEQ + row0) * 64 + (col1 & 63)] = e01;
      D[((size_t)(col0 >> 6) * T_SEQ + row1) * 64 + (col0 & 63)] = e10;
      D[((size_t)(col1 >> 6) * T_SEQ + row1) * 64 + (col1 & 63)] = e11;
    }
  }
}

// ---------------------------------------------------------------------------
// Flash attention (causal, GQA). Grid: (T/256 query tiles, 16 heads).
// Block 256 = 8 waves; wave w owns query rows qtile*256 + w*32 .. +31.
// Q fragments register-cached; K/V key blocks of 32 staged in LDS; P tile
// round-trips through per-wave LDS to convert C-layout -> A-layout.
// ---------------------------------------------------------------------------
__global__ __launch_bounds__(256) void attn_kernel(
    const float* __restrict__ Q,   // [T, DIM]
    const float* __restrict__ Kc,  // [Hk, T, 64]
    const float* __restrict__ Vc,  // [Hk, T, 64]
    float* __restrict__ Att) {     // [T, DIM]
  __shared__ float Ks[32][68];
  __shared__ float Vs[32][68];
  __shared__ float Ps[8][32][34];

  const int tid   = threadIdx.x;
  const int lane  = tid & 31;
  const int wv    = tid >> 5;
  const int l15   = lane & 15;
  const int hoff  = (lane >> 4) << 1;
  const int half8 = (lane >> 4) << 3;

  const int h     = blockIdx.y;
  const int kvh   = h >> 2;  // 4 query heads per kv head
  const int qrow0 = blockIdx.x * 256 + wv * 32;

  // Register-cache Q fragments for this wave's 32 rows (A-layout, K=64).
  v2f qa[2][16];
#pragma unroll
  for (int mt = 0; mt < 2; ++mt) {
    const float* qp = Q + (size_t)(qrow0 + mt * 16 + l15) * DIM_ + h * HD_ + hoff;
#pragma unroll
    for (int ks = 0; ks < 16; ++ks) qa[mt][ks] = *(const v2f*)(qp + ks * 4);
  }

  v8f o[2][4];
  float mi[2][8], li[2][8];
#pragma unroll
  for (int mt = 0; mt < 2; ++mt) {
#pragma unroll
    for (int nt = 0; nt < 4; ++nt) o[mt][nt] = zero8();
#pragma unroll
    for (int r = 0; r < 8; ++r) { mi[mt][r] = -3.0e38f; li[mt][r] = 0.f; }
  }

  const float* kbase = Kc + (size_t)kvh * T_SEQ * HD_;
  const float* vbase = Vc + (size_t)kvh * T_SEQ * HD_;
  const int nblk = blockIdx.x * 8 + 8;  // 32-key blocks covering causal extent
  const int lr = tid >> 3;              // K/V tile fill: 8 thr/row, 8 floats
  const int lc = (tid & 7) << 3;

  for (int blk = 0; blk < nblk; ++blk) {
    const int s0 = blk * 32;
    __syncthreads();
    {
      const float* kp = kbase + (size_t)(s0 + lr) * HD_ + lc;
      const float* vp = vbase + (size_t)(s0 + lr) * HD_ + lc;
      *(float4*)&Ks[lr][lc]     = *(const float4*)(kp);
      *(float4*)&Ks[lr][lc + 4] = *(const float4*)(kp + 4);
      *(float4*)&Vs[lr][lc]     = *(const float4*)(vp);
      *(float4*)&Vs[lr][lc + 4] = *(const float4*)(vp + 4);
    }
    __syncthreads();
    if (s0 > qrow0 + 31) continue;  // wave-uniform: fully above the diagonal

    // ---- S = Q K^T (32 x 32), contraction over hd=64 ----
    v8f sA[2][2];
    sA[0][0] = zero8(); sA[0][1] = zero8();
    sA[1][0] = zero8(); sA[1][1] = zero8();
#pragma unroll
    for (int ks = 0; ks < 16; ++ks) {
      v2f b0 = *(const v2f*)&Ks[l15][ks * 4 + hoff];
      v2f b1 = *(const v2f*)&Ks[16 + l15][ks * 4 + hoff];
      sA[0][0] = wmma_f32(qa[0][ks], b0, sA[0][0]);
      sA[0][1] = wmma_f32(qa[0][ks], b1, sA[0][1]);
      sA[1][0] = wmma_f32(qa[1][ks], b0, sA[1][0]);
      sA[1][1] = wmma_f32(qa[1][ks], b1, sA[1][1]);
    }

    // ---- scale + causal mask + online softmax ----
#pragma unroll
    for (int mt = 0; mt < 2; ++mt) {
#pragma unroll
      for (int r = 0; r < 8; ++r) {
        const int row = qrow0 + mt * 16 + half8 + r;
        float v0 = sA[mt][0][r] * SCALE_;
        float v1 = sA[mt][1][r] * SCALE_;
        if (s0 + l15 > row)      v0 = -1.0e9f;
        if (s0 + 16 + l15 > row) v1 = -1.0e9f;
        float mx = fmaxf(v0, v1);
        mx = fmaxf(mx, __shfl_xor(mx, 1, 32));
        mx = fmaxf(mx, __shfl_xor(mx, 2, 32));
        mx = fmaxf(mx, __shfl_xor(mx, 4, 32));
        mx = fmaxf(mx, __shfl_xor(mx, 8, 32));
        const float mprev = mi[mt][r];
        const float mnew  = fmaxf(mprev, mx);
        const float alpha = __expf(mprev - mnew);
        mi[mt][r] = mnew;
        const float p0 = __expf(v0 - mnew);
        const float p1 = __expf(v1 - mnew);
        float rs = p0 + p1;
        rs += __shfl_xor(rs, 1, 32);
        rs += __shfl_xor(rs, 2, 32);
        rs += __shfl_xor(rs, 4, 32);
        rs += __shfl_xor(rs, 8, 32);
        li[mt][r] = li[mt][r] * alpha + rs;
#pragma unroll
        for (int nt = 0; nt < 4; ++nt) o[mt][nt][r] *= alpha;
        Ps[wv][mt * 16 + half8 + r][l15]      = p0;
        Ps[wv][mt * 16 + half8 + r][16 + l15] = p1;
      }
    }

    // ---- O += P V (contraction over 32 keys) ----
#pragma unroll
    for (int ks = 0; ks < 32; ks += 4) {
      v2f a0 = *(const v2f*)&Ps[wv][l15][ks + hoff];
      v2f a1 = *(const v2f*)&Ps[wv][16 + l15][ks + hoff];
#pragma unroll
      for (int nt = 0; nt < 4; ++nt) {
        v2f b;
        b.x = Vs[ks + hoff][nt * 16 + l15];
        b.y = Vs[ks + hoff + 1][nt * 16 + l15];
        o[0][nt] = wmma_f32(a0, b, o[0][nt]);
        o[1][nt] = wmma_f32(a1, b, o[1][nt]);
      }
    }
  }

  // ---- finalize: divide by row sum, write [T, DIM] (head-concat layout) ----
#pragma unroll
  for (int mt = 0; mt < 2; ++mt) {
#pragma unroll
    for (int r = 0; r < 8; ++r) {
      const int row = qrow0 + mt * 16 + half8 + r;
      const float inv = 1.0f / li[mt][r];
      float* ap = Att + (size_t)row * DIM_ + h * HD_;
#pragma unroll
      for (int nt = 0; nt < 4; ++nt) ap[nt * 16 + l15] = o[mt][nt][r] * inv;
    }
  }
}

// ---------------------------------------------------------------------------
extern "C" void kernel_launch(void* const* d_in, const int* in_sizes, int n_in,
                              void* d_out, int out_size, void* d_ws,
                              size_t ws_size, hipStream_t stream) {
  (void)in_sizes; (void)n_in; (void)out_size; (void)ws_size;
  const float* x   = (const float*)d_in[0];
  // d_in[1] = causal mask (bool) — implied by the kernel, unused.
  const float* q_w = (const float*)d_in[2];
  const float* q_b = (const float*)d_in[3];
  const float* k_w = (const float*)d_in[4];
  const float* k_b = (const float*)d_in[5];
  const float* v_w = (const float*)d_in[6];
  const float* v_b = (const float*)d_in[7];
  const float* o_w = (const float*)d_in[8];
  const float* o_b = (const float*)d_in[9];

  float* out  = (float*)d_out;                        // [T, DIM]
  float* kout = out + (size_t)T_SEQ * DIM_;           // [Hk, T, 64]
  float* vout = kout + (size_t)KVH_ * T_SEQ * HD_;    // [Hk, T, 64]

  float* q_ws   = (float*)d_ws;                       // [T, DIM]
  float* att_ws = q_ws + (size_t)T_SEQ * DIM_;        // [T, DIM]

  dim3 blk(256);
  gemm_kernel<0><<<dim3(T_SEQ / 128, DIM_ / 64), blk, 0, stream>>>(
      x, q_w, q_b, q_ws, T_SEQ, DIM_, DIM_);
  gemm_kernel<1><<<dim3(T_SEQ / 128, KVDIM_ / 64), blk, 0, stream>>>(
      x, k_w, k_b, kout, T_SEQ, KVDIM_, DIM_);
  gemm_kernel<1><<<dim3(T_SEQ / 128, KVDIM_ / 64), blk, 0, stream>>>(
      x, v_w, v_b, vout, T_SEQ, KVDIM_, DIM_);
  attn_kernel<<<dim3(T_SEQ / 256, HEADS_), blk, 0, stream>>>(
      q_ws, kout, vout, att_ws);
  gemm_kernel<0><<<dim3(T_SEQ / 128, DIM_ / 64), blk, 0, stream>>>(
      att_ws, o_w, o_b, out, T_SEQ, DIM_, DIM_);
}